// NoisyCell_67456756351312
// MI455X (gfx1250) — compile-verified
//
#include <hip/hip_runtime.h>

// GRU over packed sequences: T=512, B=256, E=256, H=256.
// Persistent-kernel recurrence: 32 WGs (column-sliced), weights resident in
// LDS (96KB/WG of 320KB), h double-buffered in workspace, one grid barrier
// per timestep, fp32 WMMA (V_WMMA_F32_16X16X4_F32) for both GEMMs per step.

#define T_STEPS 512
#define B_SZ    256
#define E_SZ    256
#define H_SZ    256
#define NWG     32

typedef float v2f __attribute__((ext_vector_type(2)));
typedef float v8f __attribute__((ext_vector_type(8)));

// Fast hardware transcendentals: v_rcp_f32 / v_exp_f32 / v_tanh_f32.
__device__ __forceinline__ float sigf(float v) {
  return __builtin_amdgcn_rcpf(1.0f + __expf(-v));
}
__device__ __forceinline__ float fast_tanh(float x) {
#if __has_builtin(__builtin_amdgcn_tanhf)
  return __builtin_amdgcn_tanhf(x);
#else
  // branch-free: tanh(x) = 1 - 2/(exp(2x)+1); saturates correctly at +-inf
  const float e = __expf(2.0f * x);
  return 1.0f - 2.0f * __builtin_amdgcn_rcpf(e + 1.0f);
#endif
}

// ---------------------------------------------------------------------------
// Zero h0/h1 and the grid-barrier counter (workspace is poisoned by harness).
// ---------------------------------------------------------------------------
__global__ void __launch_bounds__(256)
gru_prep_kernel(float* __restrict__ h0, float* __restrict__ h1,
                unsigned* __restrict__ counter) {
  unsigned i = blockIdx.x * blockDim.x + threadIdx.x;
  unsigned stride = gridDim.x * blockDim.x;
  for (unsigned j = i; j < (unsigned)(B_SZ * H_SZ); j += stride) {
    h0[j] = 0.0f;
    h1[j] = 0.0f;
  }
  if (i == 0) *counter = 0u;
}

// ---------------------------------------------------------------------------
// Persistent recurrence kernel.
//   blockIdx.x = 0..31 :  ct = bid & 15  -> hidden-column tile (16 cols)
//                         mg = bid >> 4  -> which half of the 16 batch tiles
//   wave w (0..7) owns batch-row tile  mt = mg*8 + w  (16 rows).
// Per step each wave computes six 16x16 fp32 WMMA accumulations over K=256:
//   {i_r,i_z,i_n} = x_t-tile @ W_ih^T-slice,  {h_r,h_z,h_n} = h-tile @ W_hh^T.
// ---------------------------------------------------------------------------
__global__ void __launch_bounds__(256)
gru_step_kernel(const float* __restrict__ x,
                const float* __restrict__ Wih,
                const float* __restrict__ Whh,
                const float* __restrict__ bih,
                const float* __restrict__ bhh,
                const int*   __restrict__ lengths,
                float* __restrict__ h0,
                float* __restrict__ h1,
                unsigned* __restrict__ counter)
{
  // ldsW layout: [w(ih/hh)][gate][k(0..255)][c(0..15)]  = 2*3*256*16*4 = 96 KB
  extern __shared__ float ldsW[];

  const int tid  = threadIdx.x;
  const int lane = tid & 31;
  const int wave = tid >> 5;
  const int bid  = blockIdx.x;
  const int ct = bid & 15;
  const int mg = bid >> 4;
  const int mt = mg * 8 + wave;
  const int c0 = ct * 16;   // first hidden column this WG owns
  const int m0 = mt * 16;   // first batch row this wave owns

  // ---- Stage this WG's weight slice into LDS, transposed to [k][c] --------
  for (int w = 0; w < 2; ++w) {
    const float* W = w ? Whh : Wih;
    float* dst = ldsW + w * (3 * E_SZ * 16);
    for (int gc = 0; gc < 48; ++gc) {            // gate*16 + col
      const int g = gc >> 4, c = gc & 15;
      const float* src = W + (size_t)(g * H_SZ + c0 + c) * E_SZ;
      float* d2 = dst + g * (E_SZ * 16) + c;
      for (int k = tid; k < E_SZ; k += 256)      // coalesced along k
        d2[(size_t)k * 16] = src[k];
    }
  }
  __syncthreads();

  // ---- Per-lane invariants ------------------------------------------------
  const int nN = lane & 15;   // column within 16x16 tile (B/C/D row-striping)
  const int kh = lane >> 4;   // K-half for A/B fragments (ISA 16x4 layout)
  const int kb = 2 * kh;

  float bi[3], bh[3];
#pragma unroll
  for (int g = 0; g < 3; ++g) {
    bi[g] = bih[g * H_SZ + c0 + nN];
    bh[g] = bhh[g * H_SZ + c0 + nN];
  }
  int lenv[8];
#pragma unroll
  for (int v = 0; v < 8; ++v) lenv[v] = lengths[m0 + v + 8 * kh];

  const float* ldsI = ldsW;                       // W_ih slice
  const float* ldsH = ldsW + 3 * E_SZ * 16;       // W_hh slice
  const int arow = m0 + nN;                       // A-fragment row for lane

  // ---- Recurrence ---------------------------------------------------------
  for (int t = 0; t < T_STEPS; ++t) {
    const float* hprev = (t & 1) ? h1 : h0;
    float*       hnext = (t & 1) ? h0 : h1;
    const float* xrow = x + ((size_t)t * B_SZ + arow) * E_SZ;
    const float* hrow = hprev + (size_t)arow * H_SZ;

    // Warm L2 with next step's x tile while this step computes
    // (lowers to global_prefetch_b8; skipped at t = T-1, so the last
    //  step never prefetches past the x tensor).
    if (t + 1 < T_STEPS)
      __builtin_prefetch(xrow + (size_t)B_SZ * E_SZ, 0, 1);

    v8f acc[6];
#pragma unroll
    for (int i = 0; i < 6; ++i) acc[i] = {};

#pragma unroll 2
    for (int k0 = 0; k0 < E_SZ; k0 += 4) {
      // A fragments (16x4 f32): lane row = m0+(lane&15), K = k0 + 2*kh + {0,1}
      v2f ax = *(const v2f*)(xrow + k0 + kb);
      v2f ah = *(const v2f*)(hrow + k0 + kb);
#pragma unroll
      for (int g = 0; g < 3; ++g) {
        // B fragments (4x16 f32): vgpr v holds K = k0 + 2*kh + v, col = nN
        const float* pI = ldsI + (size_t)(g * E_SZ + k0 + kb) * 16 + nN;
        const float* pH = ldsH + (size_t)(g * E_SZ + k0 + kb) * 16 + nN;
        v2f bI; bI.x = pI[0]; bI.y = pI[16];
        v2f bH; bH.x = pH[0]; bH.y = pH[16];
        acc[g]     = __builtin_amdgcn_wmma_f32_16x16x4_f32(
                         false, ax, false, bI, (short)0, acc[g], false, false);
        acc[3 + g] = __builtin_amdgcn_wmma_f32_16x16x4_f32(
                         false, ah, false, bH, (short)0, acc[3 + g], false, false);
      }
    }

    // ---- Gates + masked state update (C/D layout: vgpr v -> row v + 8*kh) --
#pragma unroll
    for (int v = 0; v < 8; ++v) {
      const int row = m0 + v + 8 * kh;
      const size_t oidx = (size_t)row * H_SZ + c0 + nN;
      const float hold = hprev[oidx];
      const float r = sigf(acc[0][v] + bi[0] + acc[3][v] + bh[0]);
      const float z = sigf(acc[1][v] + bi[1] + acc[4][v] + bh[1]);
      const float n = fast_tanh(acc[2][v] + bi[2] + r * (acc[5][v] + bh[2]));
      const float hnew = (1.0f - z) * n + z * hold;
      hnext[oidx] = (t < lenv[v]) ? hnew : hold;   // packed-seq freeze
    }

    // ---- Grid-wide release/acquire barrier (double buffer -> 1/step) ------
    __syncthreads();
    if (tid == 0) {
      __threadfence();
      __hip_atomic_fetch_add(counter, 1u, __ATOMIC_RELEASE,
                             __HIP_MEMORY_SCOPE_AGENT);
      const unsigned tgt = (unsigned)NWG * (unsigned)(t + 1);
      while (__hip_atomic_load(counter, __ATOMIC_ACQUIRE,
                               __HIP_MEMORY_SCOPE_AGENT) < tgt)
        __builtin_amdgcn_s_sleep(4);
    }
    __syncthreads();
  }
}

// ---------------------------------------------------------------------------
// out[b, j] = h_final[unsorted_indices[b], j]   (final buffer is h0: T even)
// ---------------------------------------------------------------------------
__global__ void __launch_bounds__(256)
gru_gather_kernel(const float* __restrict__ hfin,
                  const int* __restrict__ idx,
                  float* __restrict__ out) {
  const int i = blockIdx.x * 256 + threadIdx.x;   // 65536 elements
  const int b = i >> 8;
  const int j = i & (H_SZ - 1);
  out[i] = hfin[(size_t)idx[b] * H_SZ + j];
}

extern "C" void kernel_launch(void* const* d_in, const int* in_sizes, int n_in,
                              void* d_out, int out_size, void* d_ws, size_t ws_size,
                              hipStream_t stream) {
  (void)in_sizes; (void)n_in; (void)out_size; (void)ws_size;
  const float* x       = (const float*)d_in[0];
  const float* Wih     = (const float*)d_in[1];
  const float* Whh     = (const float*)d_in[2];
  const float* bih     = (const float*)d_in[3];
  const float* bhh     = (const float*)d_in[4];
  const int*   lengths = (const int*)d_in[5];
  const int*   uidx    = (const int*)d_in[6];
  float*       out     = (float*)d_out;

  float*    h0      = (float*)d_ws;
  float*    h1      = h0 + (size_t)B_SZ * H_SZ;
  unsigned* counter = (unsigned*)(h1 + (size_t)B_SZ * H_SZ);

  const size_t ldsBytes = (size_t)2 * 3 * E_SZ * 16 * sizeof(float);  // 96 KB
  (void)hipFuncSetAttribute((const void*)gru_step_kernel,
                            hipFuncAttributeMaxDynamicSharedMemorySize,
                            (int)ldsBytes);

  gru_prep_kernel<<<64, 256, 0, stream>>>(h0, h1, counter);
  gru_step_kernel<<<NWG, 256, ldsBytes, stream>>>(x, Wih, Whh, bih, bhh,
                                                  lengths, h0, h1, counter);
  gru_gather_kernel<<<B_SZ * H_SZ / 256, 256, 0, stream>>>(h0, uidx, out);
}